// RealNVP_11424613007642
// MI455X (gfx1250) — compile-verified
//
#include <hip/hip_runtime.h>
#include <math.h>

// ---------------------------------------------------------------------------
// RealNVP indexed-MLP flow, fused for MI455X (gfx1250, wave32).
//
//  * Sparse formulation: dense@W1 is a weighted gather of <=64 W1 rows,
//    h@W2 only needs <=128 gathered W2 columns -> ~2 GFLOP total instead of
//    31.5 GFLOP dense + 320MB scatter traffic.
//  * All weights (15 MB incl. transposed W2) fit in the 192 MB L2 -> the
//    kernel is L2-gather bound (~2 GB of L2 reads), HBM traffic ~8 MB.
//  * Batch rows are independent -> one wave32 per row runs all 10 layers
//    fused; no inter-layer launches, no global sync.
//  * Phase B uses v_wmma_f32_16x16x4_f32: A = 16 gathered W2T rows loaded
//    directly in WMMA A layout via one global b64 gather per lane per K-step,
//    B = h replicated across N. Tile loop unrolled x2 -> two independent
//    WMMA accumulator chains + 32 in-flight gathers per wave.
//  * Phase A is branchless (clamped index, zeroed multiplier) so the
//    compiler can unroll and clause the coalesced W1-row gathers.
//  * Wave-local LDS ordering via s_wait_dscnt (CDNA5 split counters).
// ---------------------------------------------------------------------------

#define NTIPS   64
#define EMB     2000
#define HID     64
#define NLAYERS 10
#define NBRANCH 125          // 2*NTIPS-3
#define W2C     (2*EMB)      // 4000
#define WAVES   8            // waves (rows) per block

typedef float v2f __attribute__((ext_vector_type(2)));
typedef float v8f __attribute__((ext_vector_type(8)));

static __device__ __forceinline__ v8f wmma4(v2f a, v2f b, v8f c) {
  // D = A(16x4) * B(4x16) + C   (f32 WMMA, wave32)
  return __builtin_amdgcn_wmma_f32_16x16x4_f32(
      /*neg_a=*/false, a, /*neg_b=*/false, b,
      /*c_mod=*/(short)0, c, /*reuse_a=*/false, /*reuse_b=*/false);
}

// Order LDS traffic within a wave (each wave owns a private LDS slice, so a
// full __syncthreads() is unnecessary). s_wait_dscnt 0 drains DS ops; the
// memory clobber + wave barriers stop the compiler from reordering around it.
static __device__ __forceinline__ void wave_lds_fence() {
  __builtin_amdgcn_wave_barrier();
  asm volatile("s_wait_dscnt 0" ::: "memory");
  __builtin_amdgcn_wave_barrier();
}

static __device__ __forceinline__ float eluf(float x) {
  return x > 0.0f ? x : expm1f(x);
}

// ---------------------------------------------------------------------------
// Prepass: W2 (NLAYERS,64,4000) -> W2T (NLAYERS,4000,64) so each gathered
// column is a contiguous 256B row (b64-friendly gathers from L2).
// ---------------------------------------------------------------------------
__global__ void w2_transpose(const float* __restrict__ W2,
                             float* __restrict__ W2T) {
  int i = blockIdx.x * blockDim.x + threadIdx.x;
  const int per_layer = HID * W2C;
  if (i >= NLAYERS * per_layer) return;
  int l   = i / per_layer;
  int rem = i - l * per_layer;
  int k   = rem / W2C;        // 0..63
  int c   = rem - k * W2C;    // 0..3999  (contiguous across threads -> coalesced read)
  W2T[(size_t)l * per_layer + (size_t)c * HID + k] = W2[i];
}

// ---------------------------------------------------------------------------
// Fused flow kernel: one wave32 per batch row, all NLAYERS layers.
// TRANS: W2x is the (4000,64)-per-layer transposed weight (fast path).
// ---------------------------------------------------------------------------
template <bool TRANS>
__global__ __launch_bounds__(WAVES * 32)
void realnvp_fused(const float* __restrict__ samp,
                   const float* __restrict__ logq_in,
                   const float* __restrict__ W1,
                   const float* __restrict__ b1,
                   const float* __restrict__ W2x,
                   const float* __restrict__ b2,
                   const int*   __restrict__ nidx,
                   float* __restrict__ out,
                   int Bn) {
  __shared__ float s_val[WAVES][2][64];                 // side values
  __shared__ int   s_idx[WAVES][2][64];                 // side indices (static)
  __shared__ __align__(16) float s_h[WAVES][64];        // hidden activations
  __shared__ float s_stage[WAVES][128];                 // raw s (0..63) / t (64..127)

  const int lane = threadIdx.x & 31;
  const int w    = threadIdx.x >> 5;
  const int r    = blockIdx.x * WAVES + w;
  if (r >= Bn) return;

  const int n1 = NBRANCH - NTIPS;                       // 61
  const int nside[2] = {NTIPS, n1};                     // side0=64, side1=61

  // ---- load row state into this wave's LDS slice ----
  const float* srow = samp + (size_t)r * NBRANCH;
  const int*   irow = nidx + (size_t)r * NBRANCH;
  s_val[w][0][lane]      = srow[lane];
  s_val[w][0][lane + 32] = srow[lane + 32];
  s_idx[w][0][lane]      = irow[lane];
  s_idx[w][0][lane + 32] = irow[lane + 32];
#pragma unroll
  for (int q = 0; q < 2; ++q) {
    int j = lane + q * 32;
    if (j < n1) {
      s_val[w][1][j] = srow[NTIPS + j];
      s_idx[w][1][j] = irow[NTIPS + j];
    } else {                                            // pad slots 61..63
      s_val[w][1][j] = 0.0f;
      s_idx[w][1][j] = EMB;
    }
  }
  wave_lds_fence();

  const int off2 = (lane >> 4) << 1;                    // 0 (lanes 0-15) / 2 (16-31)
  float lq = 0.0f;                                      // sum log(s) accumulator

  for (int i = 0; i < NLAYERS; ++i) {
    const int inSide  = (i & 1) ? 0 : 1;                // layer0 consumes side1 (=b)
    const int outSide = 1 - inSide;
    const int n_in    = nside[inSide];
    const int n_out   = nside[outSide];

    const float* W1l = W1  + (size_t)i * EMB * HID;
    const float* b1l = b1  + (size_t)i * HID;
    const float* W2l = W2x + (size_t)i * HID * W2C;     // same size either layout
    const float* b2l = b2  + (size_t)i * W2C;

    // ---- Phase A: h = elu( sum_j x[j] * W1[idx[j],:] + b1 ) ----
    // Branchless: idx==PAD contributes 0 via zeroed multiplier; index clamped
    // to row 0 so the loads stay unconditional and the compiler can unroll
    // and clause-batch the coalesced 256B W1-row gathers.
    float acc0 = b1l[lane];
    float acc1 = b1l[lane + 32];
#pragma unroll 4
    for (int j = 0; j < n_in; ++j) {
      float x   = s_val[w][inSide][j];                  // uniform LDS broadcast
      int   id  = s_idx[w][inSide][j];
      int   idc = (id < EMB) ? id : 0;
      float xm  = (id < EMB) ? x : 0.0f;
      const float* wr = W1l + (size_t)idc * HID;
      acc0 = fmaf(xm, wr[lane], acc0);
      acc1 = fmaf(xm, wr[lane + 32], acc1);
    }
    s_h[w][lane]      = eluf(acc0);
    s_h[w][lane + 32] = eluf(acc1);
    wave_lds_fence();

    // ---- B operand: h replicated across N, in 16x16x4 B layout ----
    v2f Bv[16];
#pragma unroll
    for (int kk = 0; kk < 16; ++kk)
      Bv[kk] = *(const v2f*)&s_h[w][kk * 4 + off2];

    // ---- Phase B: 8 output tiles (4 s-tiles, 4 t-tiles) of 16 gathered
    // dot products each; x2 unroll -> two independent WMMA chains. ----
    const int m16 = lane & 15;
#pragma unroll 2
    for (int t = 0; t < 8; ++t) {
      const bool isT = (t >= 4);
      const int  m   = (t & 3) * 16 + m16;
      int c   = (m < n_out) ? s_idx[w][outSide][m] : 0;
      int row = isT ? ((c < EMB) ? c + EMB : 0) : c;    // t-col; dummy row if idx==PAD

      v8f C = {};
      if (TRANS) {
        const float* abase = W2l + (size_t)row * HID + off2;
#pragma unroll
        for (int kk = 0; kk < 16; ++kk) {
          v2f A = *(const v2f*)(abase + kk * 4);        // b64 gather, WMMA A layout
          C = wmma4(A, Bv[kk], C);
        }
      } else {                                          // untransposed W2 fallback
#pragma unroll
        for (int kk = 0; kk < 16; ++kk) {
          v2f A;
          A.x = W2l[(size_t)(kk * 4 + off2)     * W2C + row];
          A.y = W2l[(size_t)(kk * 4 + off2 + 1) * W2C + row];
          C = wmma4(A, Bv[kk], C);
        }
      }
      // D[m][*] identical across N; VGPR v holds M=v (lanes 0-15), M=v+8 (16-31)
      if (m16 == 0) {
        float* dst = &s_stage[w][(isT ? 64 : 0) + (t & 3) * 16 + (lane >> 4) * 8];
#pragma unroll
        for (int v = 0; v < 8; ++v) dst[v] = C[v];
      }
    }
    wave_lds_fence();

    // ---- affine update + logq ----
#pragma unroll
    for (int q = 0; q < 2; ++q) {
      int m = lane + q * 32;
      if (m < n_out) {
        int   c  = s_idx[w][outSide][m];
        float sr = s_stage[w][m] + b2l[c] + 2.0f;
        float sg = 1.0f / (1.0f + expf(-sr));
        float tv = (c < EMB) ? (s_stage[w][64 + m] + b2l[c + EMB]) : 0.0f;
        s_val[w][outSide][m] = sg * s_val[w][outSide][m] + tv;
        lq += logf(sg);
      }
    }
    wave_lds_fence();
  }

  // ---- outputs: samp' = concat(side0, side1) - 2 ; logq' = logq - sum log s ----
  float* orow = out + (size_t)r * NBRANCH;
  orow[lane]      = s_val[w][0][lane]      - 2.0f;
  orow[lane + 32] = s_val[w][0][lane + 32] - 2.0f;
  if (lane < n1)      orow[NTIPS + lane]      = s_val[w][1][lane]      - 2.0f;
  if (lane + 32 < n1) orow[NTIPS + lane + 32] = s_val[w][1][lane + 32] - 2.0f;

#pragma unroll
  for (int off = 16; off > 0; off >>= 1)
    lq += __shfl_xor(lq, off, 32);
  if (lane == 0)
    out[(size_t)Bn * NBRANCH + r] = logq_in[r] - lq;
}

// ---------------------------------------------------------------------------
extern "C" void kernel_launch(void* const* d_in, const int* in_sizes, int n_in,
                              void* d_out, int out_size, void* d_ws, size_t ws_size,
                              hipStream_t stream) {
  const float* samp = (const float*)d_in[0];
  const float* logq = (const float*)d_in[1];
  const float* W1   = (const float*)d_in[2];
  const float* b1   = (const float*)d_in[3];
  const float* W2   = (const float*)d_in[4];
  const float* b2   = (const float*)d_in[5];
  const int*   idx  = (const int*)d_in[6];
  float* out = (float*)d_out;

  const int Bn = in_sizes[1];                           // logq_branch is (B,)
  const int grid = (Bn + WAVES - 1) / WAVES;
  const size_t needT = (size_t)NLAYERS * W2C * HID * sizeof(float);  // 10.24 MB

  if (ws_size >= needT) {
    float* W2T = (float*)d_ws;
    const int total = NLAYERS * HID * W2C;
    w2_transpose<<<(total + 255) / 256, 256, 0, stream>>>(W2, W2T);
    realnvp_fused<true><<<grid, WAVES * 32, 0, stream>>>(
        samp, logq, W1, b1, W2T, b2, idx, out, Bn);
  } else {
    realnvp_fused<false><<<grid, WAVES * 32, 0, stream>>>(
        samp, logq, W1, b1, W2, b2, idx, out, Bn);
  }
}